// GAU_74534862455342
// MI455X (gfx1250) — compile-verified
//
#include <hip/hip_runtime.h>
#include <math.h>

#define EMBED_DIM 128
#define HIDDEN_DIM 256
#define ROT_DIM 32
#define SEQ 4096
#define BATCH 4

typedef float v2f __attribute__((ext_vector_type(2)));
typedef float v8f __attribute__((ext_vector_type(8)));

__device__ __forceinline__ v8f wmma4(v2f a, v2f b, v8f c) {
  // D = A(16x4, f32) * B(4x16, f32) + C(16x16, f32)
  return __builtin_amdgcn_wmma_f32_16x16x4_f32(false, a, false, b,
                                               (short)0, c, false, false);
}

__device__ __forceinline__ v8f zero8() {
  v8f z = {0.f, 0.f, 0.f, 0.f, 0.f, 0.f, 0.f, 0.f};
  return z;
}

// ---------------------------------------------------------------------------
// Kernel 1: RoPE + LayerNorm, one block (128 threads) per row.
// ---------------------------------------------------------------------------
__global__ __launch_bounds__(128) void k_rope_ln(
    const float* __restrict__ query, const float* __restrict__ ln_g,
    const float* __restrict__ ln_b, float* __restrict__ normed) {
  const int row  = blockIdx.x;          // 0 .. B*S-1
  const int s    = row & (SEQ - 1);     // position within sequence
  const int d    = threadIdx.x;         // channel 0..127
  const int wave = d >> 5;
  const int lane = d & 31;
  const float* x = query + (size_t)row * EMBED_DIM;

  float v = x[d];
  if (d < ROT_DIM) {
    int i = d >> 1;
    float invf = 1.0f / powf(10000.0f, (float)(2 * i) / 32.0f);
    float ang  = (float)s * invf;
    float c = cosf(ang), sn = sinf(ang);
    float other = (d & 1) ? x[d - 1] : -x[d + 1];
    v = v * c + other * sn;
  }

  __shared__ float partial[4];
  // mean
  float r0 = v;
#pragma unroll
  for (int off = 16; off > 0; off >>= 1) r0 += __shfl_xor(r0, off, 32);
  if (lane == 0) partial[wave] = r0;
  __syncthreads();
  float mu = (partial[0] + partial[1] + partial[2] + partial[3]) * (1.0f / 128.0f);
  __syncthreads();
  // variance
  float dv = v - mu;
  float r1 = dv * dv;
#pragma unroll
  for (int off = 16; off > 0; off >>= 1) r1 += __shfl_xor(r1, off, 32);
  if (lane == 0) partial[wave] = r1;
  __syncthreads();
  float var = (partial[0] + partial[1] + partial[2] + partial[3]) * (1.0f / 128.0f);
  float rs = rsqrtf(var + 1e-5f);
  normed[(size_t)row * EMBED_DIM + d] = dv * rs * ln_g[d] + ln_b[d];
}

// ---------------------------------------------------------------------------
// GEMM tile helper: C(16x16) = A(16x128, LDS) @ W(128x16 slice of row-major
// [128 x LD] global).  LD is compile-time so W loads become immediate-offset
// global_load_b32 off one per-lane base pointer.
// ---------------------------------------------------------------------------
template <int LD>
__device__ __forceinline__ v8f gemm_tile_k128(const float (*A)[132],
                                              const float* __restrict__ W,
                                              int n0, int m, int kb_off) {
  const float* Wp = W + kb_off * LD + n0 + m;  // per-lane base
  v8f acc = zero8();
#pragma unroll
  for (int s4 = 0; s4 < 32; ++s4) {
    v2f a; a.x = A[m][s4 * 4 + kb_off]; a.y = A[m][s4 * 4 + kb_off + 1];
    v2f b; b.x = Wp[s4 * 4 * LD];       b.y = Wp[s4 * 4 * LD + LD];
    acc = wmma4(a, b, acc);
  }
  return acc;
}

// ---------------------------------------------------------------------------
// Kernel 2: gate = silu(normed@Wg+bg), v = silu(value@Wg+bg),
//           qk = silu(normed@Wqk+bqk) -> q,k affine.  WMMA f32 16x16x4.
// One block = 128 threads (4 waves) handles a 16-row band; 40 col-tiles.
// ---------------------------------------------------------------------------
__global__ __launch_bounds__(128) void k_proj(
    const float* __restrict__ normed, const float* __restrict__ value,
    const float* __restrict__ W_gate, const float* __restrict__ b_gate,
    const float* __restrict__ W_qk, const float* __restrict__ b_qk,
    const float* __restrict__ os_gamma, const float* __restrict__ os_beta,
    float* __restrict__ gate, float* __restrict__ vbuf,
    float* __restrict__ qbuf, float* __restrict__ kbuf) {
  const int row0 = blockIdx.x * 16;
  const int tid  = threadIdx.x;
  const int wave = tid >> 5;
  const int lane = tid & 31;
  const int m    = lane & 15;
  const int hi   = lane >> 4;
  const int kb_off = hi * 2;

  __shared__ float nA[16][132];
  __shared__ float vA[16][132];
  // vectorized staging: 512 float4 per matrix
  for (int e = tid; e < 512; e += 128) {
    int r = e >> 5, c = (e * 4) & 127;
    *(float4*)&nA[r][c] = *(const float4*)&normed[(size_t)(row0 + r) * EMBED_DIM + c];
    *(float4*)&vA[r][c] = *(const float4*)&value[(size_t)(row0 + r) * EMBED_DIM + c];
  }
  __syncthreads();

  for (int t = wave; t < 40; t += 4) {
    const bool is_v  = (t >= 16) && (t < 32);
    const bool is_qk = (t >= 32);
    const int  n0    = is_qk ? (t - 32) * 16 : (is_v ? (t - 16) * 16 : t * 16);

    v8f acc;
    if (is_qk)      acc = gemm_tile_k128<EMBED_DIM>(nA, W_qk, n0, m, kb_off);
    else if (is_v)  acc = gemm_tile_k128<HIDDEN_DIM>(vA, W_gate, n0, m, kb_off);
    else            acc = gemm_tile_k128<HIDDEN_DIM>(nA, W_gate, n0, m, kb_off);

    const int n = n0 + m;
    if (!is_qk) {
      float bias = b_gate[n];
      float* dst = is_v ? vbuf : gate;
#pragma unroll
      for (int r = 0; r < 8; ++r) {
        int M = r + hi * 8;
        float x = acc[r] + bias;
        float sl = x / (1.0f + expf(-x));
        dst[(size_t)(row0 + M) * HIDDEN_DIM + n] = sl;
      }
    } else {
      float bias = b_qk[n];
      float g0 = os_gamma[n], g1 = os_gamma[EMBED_DIM + n];
      float e0 = os_beta[n],  e1 = os_beta[EMBED_DIM + n];
#pragma unroll
      for (int r = 0; r < 8; ++r) {
        int M = r + hi * 8;
        float x = acc[r] + bias;
        float sl = x / (1.0f + expf(-x));
        qbuf[(size_t)(row0 + M) * EMBED_DIM + n] = sl * g0 + e0;
        kbuf[(size_t)(row0 + M) * EMBED_DIM + n] = sl * g1 + e1;
      }
    }
  }
}

// ---------------------------------------------------------------------------
// Kernel 3: fused attention. Block = 4 waves, owns 64 query rows (16/wave).
// Per 16-wide j-tile: sim = q@k^T (WMMA, K=128), relu^2 -> att_map (NT store)
// + per-wave LDS stage; accumulate attn@v into 16x256 registers using a
// TRANSPOSED v tile so B-fragments are single ds_load_b64.
// Epilogue: (*gate) @ W_out + b_out via 16-col staging chunks; W_out loads
// are immediate-offset off one per-lane base.
// ---------------------------------------------------------------------------
__global__ __launch_bounds__(128) void k_attn(
    const float* __restrict__ qbuf, const float* __restrict__ kbuf,
    const float* __restrict__ vbuf, const float* __restrict__ gate,
    const float* __restrict__ W_out, const float* __restrict__ b_out,
    float* __restrict__ out, float* __restrict__ att_map) {
  const int bidx = blockIdx.x >> 6;         // batch (64 blocks per batch)
  const int blk  = blockIdx.x & 63;
  const int row0 = blk * 64;                // first query row (within batch)
  const int tid  = threadIdx.x;
  const int wave = tid >> 5;
  const int lane = tid & 31;
  const int m    = lane & 15;
  const int hi   = lane >> 4;
  const int kb_off = hi * 2;
  const int wrow0  = row0 + wave * 16;      // this wave's 16 rows

  const size_t bq = (size_t)bidx * SEQ * EMBED_DIM;
  const size_t bh = (size_t)bidx * SEQ * HIDDEN_DIM;

  __shared__ float qS[64][132];   // 33,792 B
  __shared__ float kS[16][132];   //  8,448 B
  __shared__ float vT[256][18];   // 18,432 B (transposed v tile: vT[n][k])
  __shared__ float aS[4][16][18]; //  4,608 B   -> total 65,280 B

  for (int e = tid; e < 2048; e += 128) {   // 64x128 floats as float4
    int r = e >> 5, c = (e * 4) & 127;
    *(float4*)&qS[r][c] = *(const float4*)&qbuf[bq + (size_t)(row0 + r) * EMBED_DIM + c];
  }

  v8f oacc[16];
#pragma unroll
  for (int i = 0; i < 16; ++i) oacc[i] = zero8();

  const float inv_s = 1.0f / (float)SEQ;

  for (int j0 = 0; j0 < SEQ; j0 += 16) {
    __syncthreads();  // protect kS/vT from previous iteration's readers
    for (int e = tid; e < 512; e += 128) {  // k tile 16x128 as float4
      int r = e >> 5, c = (e * 4) & 127;
      *(float4*)&kS[r][c] = *(const float4*)&kbuf[bq + (size_t)(j0 + r) * EMBED_DIM + c];
    }
    for (int e = tid; e < 1024; e += 128) { // v tile 16x256, transposed store
      int idx = e * 4;
      int r = idx >> 8, c = idx & 255;
      float4 f = *(const float4*)&vbuf[bh + (size_t)(j0 + r) * HIDDEN_DIM + c];
      vT[c + 0][r] = f.x; vT[c + 1][r] = f.y; vT[c + 2][r] = f.z; vT[c + 3][r] = f.w;
    }
    __syncthreads();

    // sim = q (16x128) @ k^T (128x16)
    v8f sim = zero8();
#pragma unroll
    for (int s4 = 0; s4 < 32; ++s4) {
      int kk = s4 * 4 + kb_off;
      v2f a; a.x = qS[wave * 16 + m][kk]; a.y = qS[wave * 16 + m][kk + 1];
      v2f b; b.x = kS[m][kk];             b.y = kS[m][kk + 1];
      sim = wmma4(a, b, sim);
    }

    // relu^2, stream to att_map, stage for A-layout reuse
#pragma unroll
    for (int r = 0; r < 8; ++r) {
      int M = r + hi * 8;
      float x = sim[r] * inv_s;
      x = (x > 0.f) ? x * x : 0.f;
      size_t idx = ((size_t)bidx * SEQ + (size_t)(wrow0 + M)) * SEQ + j0 + m;
      __builtin_nontemporal_store(x, &att_map[idx]);
      aS[wave][M][m] = x;
    }
    __builtin_amdgcn_wave_barrier();  // keep stage-stores before stage-loads

    // out += attn (16x16) @ v (16x256);  B-frag = one b64 from vT
#pragma unroll
    for (int nt = 0; nt < 16; ++nt) {
      int n0 = nt * 16;
      v8f c = oacc[nt];
#pragma unroll
      for (int s4 = 0; s4 < 4; ++s4) {
        int kk = s4 * 4 + kb_off;
        v2f a; a.x = aS[wave][m][kk];   a.y = aS[wave][m][kk + 1];
        v2f b; b.x = vT[n0 + m][kk];    b.y = vT[n0 + m][kk + 1];
        c = wmma4(a, b, c);
      }
      oacc[nt] = c;
    }
  }

  // Epilogue: out2 = (oacc * gate) @ W_out + b_out  (K = 256, 16-col chunks)
  v8f o2[8];
#pragma unroll
  for (int i = 0; i < 8; ++i) o2[i] = zero8();

  const float* WoBase = W_out + kb_off * EMBED_DIM + m;  // per-lane base

#pragma unroll
  for (int kc = 0; kc < 16; ++kc) {
#pragma unroll
    for (int r = 0; r < 8; ++r) {
      int M = r + hi * 8;
      float g = gate[bh + (size_t)(wrow0 + M) * HIDDEN_DIM + kc * 16 + m];
      aS[wave][M][m] = oacc[kc][r] * g;
    }
    __builtin_amdgcn_wave_barrier();
#pragma unroll
    for (int nt = 0; nt < 8; ++nt) {
      int n0 = nt * 16;
      v8f c = o2[nt];
#pragma unroll
      for (int s4 = 0; s4 < 4; ++s4) {
        int kk = s4 * 4 + kb_off;               // lane part already in WoBase
        v2f a; a.x = aS[wave][m][kk];           a.y = aS[wave][m][kk + 1];
        v2f b;
        b.x = WoBase[(kc * 16 + s4 * 4) * EMBED_DIM + n0];
        b.y = WoBase[(kc * 16 + s4 * 4) * EMBED_DIM + EMBED_DIM + n0];
        c = wmma4(a, b, c);
      }
      o2[nt] = c;
    }
    __builtin_amdgcn_wave_barrier();
  }

#pragma unroll
  for (int nt = 0; nt < 8; ++nt) {
    int n = nt * 16 + m;
    float bias = b_out[n];
#pragma unroll
    for (int r = 0; r < 8; ++r) {
      int M = r + hi * 8;
      out[((size_t)bidx * SEQ + (size_t)(wrow0 + M)) * EMBED_DIM + n] = o2[nt][r] + bias;
    }
  }
}

// ---------------------------------------------------------------------------
extern "C" void kernel_launch(void* const* d_in, const int* in_sizes, int n_in,
                              void* d_out, int out_size, void* d_ws, size_t ws_size,
                              hipStream_t stream) {
  (void)in_sizes; (void)n_in; (void)out_size; (void)ws_size;
  const float* query    = (const float*)d_in[0];
  // d_in[1] = 'key' — unused by the reference
  const float* value    = (const float*)d_in[2];
  const float* ln_g     = (const float*)d_in[3];
  const float* ln_b     = (const float*)d_in[4];
  const float* W_gate   = (const float*)d_in[5];
  const float* b_gate   = (const float*)d_in[6];
  const float* W_qk     = (const float*)d_in[7];
  const float* b_qk     = (const float*)d_in[8];
  const float* os_gamma = (const float*)d_in[9];
  const float* os_beta  = (const float*)d_in[10];
  const float* W_out    = (const float*)d_in[11];
  const float* b_out    = (const float*)d_in[12];

  const size_t ND = (size_t)BATCH * SEQ * EMBED_DIM;   // 2,097,152
  const size_t NH = (size_t)BATCH * SEQ * HIDDEN_DIM;  // 4,194,304

  float* out = (float*)d_out;          // first tuple element
  float* att = out + ND;               // second tuple element (B,S,S)

  float* ws     = (float*)d_ws;        // 14,680,064 floats = 56 MiB scratch
  float* normed = ws;
  float* qb     = normed + ND;
  float* kb     = qb + ND;
  float* gt     = kb + ND;
  float* vb     = gt + NH;

  k_rope_ln<<<BATCH * SEQ, 128, 0, stream>>>(query, ln_g, ln_b, normed);
  k_proj<<<BATCH * SEQ / 16, 128, 0, stream>>>(normed, value, W_gate, b_gate,
                                               W_qk, b_qk, os_gamma, os_beta,
                                               gt, vb, qb, kb);
  k_attn<<<BATCH * (SEQ / 64), 128, 0, stream>>>(qb, kb, vb, gt, W_out, b_out,
                                                 out, att);
}